// VectorQuantizer_gcn_47072841564917
// MI455X (gfx1250) — compile-verified
//
#include <hip/hip_runtime.h>
#include <hip/hip_bf16.h>
#include <cstdint>

typedef float v2f __attribute__((ext_vector_type(2)));
typedef float v8f __attribute__((ext_vector_type(8)));
typedef unsigned int v4u __attribute__((ext_vector_type(4)));
typedef int v4i __attribute__((ext_vector_type(4)));
typedef int v8i __attribute__((ext_vector_type(8)));

#define EMB_DIM 192
#define NUM_EMB 50
#define NPAD 64                         // padded embedding count (4 col-tiles of 16)
#define KSTEPS 48                       // 192 / 4
#define BFRAG_FLOATS (KSTEPS * 4 * 64)  // 12288 floats = 48KB swizzled B
#define ENORM_OFF BFRAG_FLOATS          // 12288
#define PART_OFF (BFRAG_FLOATS + NPAD)  // 12352
#define MAIN_BLOCKS 512
#define MAIN_THREADS 256
#define WAVES_PER_BLOCK (MAIN_THREADS / 32)
#define NPART (MAIN_BLOCKS * WAVES_PER_BLOCK)   // 4096

#define TILE_DW (16 * EMB_DIM)          // 3072 dwords per X tile
// TDM pads LDS dest by 2 dwords every 64 dwords -> row stride 198 dwords
// (198 mod 64 banks = 6 -> conflict-free half-wave b64 fragment reads)
#define ABUF_DW (TILE_DW + 2 * (TILE_DW / 64))  // 3168 dwords = 12672 B

// ---------------------------------------------------------------------------
// TDM: DMA one contiguous 12KB X tile into LDS with padding (ISA ch.8 D#).
// ---------------------------------------------------------------------------
__device__ __forceinline__ void tdm_load_tile(const float* X, int tile, unsigned ldsAddr) {
    unsigned long long gaddr =
        (unsigned long long)(uintptr_t)X + (unsigned long long)tile * (TILE_DW * 4ull);
    v4u g0;
    g0.x = 1u;                                   // count=1, user descriptor, no gather
    g0.y = ldsAddr;                              // LDS byte address of tile start
    g0.z = (unsigned)(gaddr & 0xFFFFFFFFu);      // global_addr[31:0]
    g0.w = (unsigned)((gaddr >> 32) & 0x01FFFFFFu) | 0x80000000u;  // addr[56:32] | type=2
    v8i g1;
    g1[0] = (int)((2u << 16)      // data_size = 4 bytes
                | (1u << 20)      // pad_enable
                | (5u << 22)      // pad_interval: every 64 dwords
                | (1u << 25));    // pad_amount: 2 dwords
    g1[1] = (int)((unsigned)TILE_DW << 16);   // tensor_dim0[15:0]
    g1[2] = (int)(1u << 16);                  // tensor_dim0 hi=0, tensor_dim1=1
    g1[3] = (int)((unsigned)TILE_DW << 16);   // tensor_dim1 hi=0, tile_dim0=3072
    g1[4] = 1;                                // tile_dim1=1, tile_dim2=0
    g1[5] = TILE_DW;                          // tensor_dim0_stride[31:0]
    g1[6] = 0;
    g1[7] = 0;
    v4i z4 = {0, 0, 0, 0};
    v8i z8 = {0, 0, 0, 0, 0, 0, 0, 0};
    __builtin_amdgcn_tensor_load_to_lds(g0, g1, z4, z4, z8, 0);
}

// ---------------------------------------------------------------------------
// Setup: WMMA-B fragment layout of padded embed^T + column squared norms.
// Fragment flat index: ((kk*4 + ct)*32 + lane)*2 + f
//   lane holds col = ct*16 + (lane&15), k = kk*4 + (lane>>4)*2 + f
// ---------------------------------------------------------------------------
__global__ void vq_setup_kernel(const float* __restrict__ ew, float* __restrict__ ws) {
    int tid = threadIdx.x;
    for (int flat = tid; flat < BFRAG_FLOATS; flat += blockDim.x) {
        int f  = flat & 1;
        int L  = (flat >> 1) & 31;
        int ct = (flat >> 6) & 3;
        int kk = flat >> 8;
        int col = ct * 16 + (L & 15);
        int k   = kk * 4 + ((L >> 4) << 1) + f;
        ws[flat] = (col < NUM_EMB) ? ew[col * EMB_DIM + k] : 0.0f;
    }
    for (int c = tid; c < NPAD; c += blockDim.x) {
        float s;
        if (c < NUM_EMB) {
            s = 0.0f;
            for (int k = 0; k < EMB_DIM; ++k) {
                float e = ew[c * EMB_DIM + k];
                s = fmaf(e, e, s);
            }
        } else {
            s = 3.0e38f;   // padded columns can never win the argmin
        }
        ws[ENORM_OFF + c] = s;
    }
}

// ---------------------------------------------------------------------------
// Main: TDM double-buffers 16-row X tiles into LDS; S(16x64) accumulated with
// v_wmma_f32_16x16x4_f32; per-row argmin of |e|^2 - 2*S; write quantized rows,
// indices, and deterministic per-wave loss partials.
// ---------------------------------------------------------------------------
__global__ __launch_bounds__(MAIN_THREADS)
void vq_main_kernel(const float* __restrict__ X, const float* __restrict__ EW,
                    const float* __restrict__ ws, float* __restrict__ out,
                    float* __restrict__ partials, int nrows) {
    __shared__ __align__(16) float Bsh[BFRAG_FLOATS];
    __shared__ __align__(16) float Ash[WAVES_PER_BLOCK * 2 * ABUF_DW];

    const int tid = threadIdx.x;
    // Cooperative copy of pre-swizzled B into LDS (b128, coalesced).
    {
        const float4* __restrict__ src = (const float4*)ws;
        float4* dst = (float4*)Bsh;
        #pragma unroll 4
        for (int i = tid; i < BFRAG_FLOATS / 4; i += MAIN_THREADS) dst[i] = src[i];
    }
    __syncthreads();

    const int lane = tid & 31;
    const int nl   = lane & 15;   // N (col) / M (row) index within 16
    const int hl   = lane >> 4;   // half-wave select
    const int waveInBlk = tid >> 5;
    const int gwave  = blockIdx.x * WAVES_PER_BLOCK + waveInBlk;
    const int nwaves = gridDim.x * WAVES_PER_BLOCK;
    const int ntiles = nrows >> 4;

    const float en0 = ws[ENORM_OFF +  0 + nl];
    const float en1 = ws[ENORM_OFF + 16 + nl];
    const float en2 = ws[ENORM_OFF + 32 + nl];
    const float en3 = ws[ENORM_OFF + 48 + nl];

    float* __restrict__ awave = &Ash[waveInBlk * (2 * ABUF_DW)];
    const unsigned aLds0 = (unsigned)(uintptr_t)(void*)awave;   // LDS byte address

    // Padded-LDS bases (pad groups are 64 tile-flat dwords; rows are 3 groups)
    const int afbase = nl * 198 + (hl << 1);   // A-fragment base for this lane

    float* __restrict__ outIdx = out + (size_t)nrows * EMB_DIM;
    float lossAcc = 0.0f;

    int tile = gwave;
    int buf = 0;
    if (tile < ntiles) tdm_load_tile(X, tile, aLds0);           // prologue prefetch

    for (; tile < ntiles; tile += nwaves, buf ^= 1) {
        const int next = tile + nwaves;
        if (next < ntiles) {
            tdm_load_tile(X, next, aLds0 + (unsigned)((buf ^ 1) * (ABUF_DW * 4)));
            __builtin_amdgcn_s_wait_tensorcnt(1);   // current tile's DMA done
        } else {
            __builtin_amdgcn_s_wait_tensorcnt(0);
        }
        __asm__ volatile("" ::: "memory");          // keep ds reads below the wait

        const float* __restrict__ abuf = awave + buf * ABUF_DW;
        const float* __restrict__ ab   = abuf + afbase;
        const float* __restrict__ bb   = Bsh + lane * 2;

        v8f c0 = {0.f,0.f,0.f,0.f,0.f,0.f,0.f,0.f};
        v8f c1 = c0, c2 = c0, c3 = c0;

        // 48 K-steps as 3 pad-groups x 16; all LDS offsets are immediates.
        #pragma unroll
        for (int k16 = 0; k16 < 3; ++k16) {
            #pragma unroll
            for (int kj = 0; kj < 16; ++kj) {
                const int aoff = k16 * 66 + kj * 4;            // 64 dw + 2 pad per group
                const int boff = (k16 * 16 + kj) * 256;        // kk*4 fragments of 64 dw
                v2f a  = *(const v2f*)(ab + aoff);
                v2f b0 = *(const v2f*)(bb + boff +   0);
                v2f b1 = *(const v2f*)(bb + boff +  64);
                v2f b2 = *(const v2f*)(bb + boff + 128);
                v2f b3 = *(const v2f*)(bb + boff + 192);
                c0 = __builtin_amdgcn_wmma_f32_16x16x4_f32(false, a, false, b0, (short)0, c0, false, false);
                c1 = __builtin_amdgcn_wmma_f32_16x16x4_f32(false, a, false, b1, (short)0, c1, false, false);
                c2 = __builtin_amdgcn_wmma_f32_16x16x4_f32(false, a, false, b2, (short)0, c2, false, false);
                c3 = __builtin_amdgcn_wmma_f32_16x16x4_f32(false, a, false, b3, (short)0, c3, false, false);
            }
        }

        // Per-row argmin. Lane holds S[row = v + 8*hl][col = ct*16 + nl].
        int bc[8];
        #pragma unroll
        for (int v = 0; v < 8; ++v) {
            float d0 = fmaf(-2.0f, c0[v], en0);
            float d1 = fmaf(-2.0f, c1[v], en1);
            float d2 = fmaf(-2.0f, c2[v], en2);
            float d3 = fmaf(-2.0f, c3[v], en3);
            float best = d0; int bcol = nl;
            if (d1 < best) { best = d1; bcol = 16 + nl; }
            if (d2 < best) { best = d2; bcol = 32 + nl; }
            if (d3 < best) { best = d3; bcol = 48 + nl; }
            // Branchless butterfly min in each 16-lane half; tie -> lowest col
            #pragma unroll
            for (int off = 1; off <= 8; off <<= 1) {
                float od = __shfl_xor(best, off, 32);
                int   oc = __shfl_xor(bcol, off, 32);
                int take = (int)(od < best) | ((int)(od == best) & (int)(oc < bcol));
                best = take ? od : best;
                bcol = take ? oc : bcol;
            }
            bc[v] = bcol;
        }

        // Emit quantized rows (+ loss) and indices; x re-read from LDS tile.
        float myIdxF = 0.0f;
        const size_t rowBase = (size_t)tile * 16;
        const float* __restrict__ xb = abuf + lane;   // padded x base for this lane
        #pragma unroll
        for (int r = 0; r < 16; ++r) {
            int bcr = __shfl(bc[r & 7], (r < 8) ? 0 : 16, 32);  // wave-uniform
            if (lane == r) myIdxF = (float)bcr;
            const float* __restrict__ qrow = EW + bcr * EMB_DIM;
            float* __restrict__ orow = out + (rowBase + (size_t)r) * EMB_DIM;
            #pragma unroll
            for (int j = 0; j < 6; ++j) {
                int colj = lane + 32 * j;
                float q = qrow[colj];
                float x = xb[r * 198 + 32 * j + (j & ~1)];   // padidx folded
                orow[colj] = x + (q - x);    // quantized_st, bit-exact formula
                float df = q - x;
                lossAcc = fmaf(df, df, lossAcc);
            }
        }
        if (lane < 16) outIdx[rowBase + lane] = myIdxF;
    }

    // Deterministic loss partial per wave (no float atomics).
    #pragma unroll
    for (int off = 1; off <= 16; off <<= 1) lossAcc += __shfl_xor(lossAcc, off, 32);
    if (lane == 0) partials[gwave] = lossAcc;
}

// ---------------------------------------------------------------------------
// Final fixed-order reduction -> commitloss, contrastloss.
// ---------------------------------------------------------------------------
__global__ void vq_reduce_kernel(const float* __restrict__ partials, float* __restrict__ out,
                                 int npart, float invTotal, unsigned long long lossOff) {
    __shared__ float sh[256];
    int tid = threadIdx.x;
    float s = 0.0f;
    for (int i = tid; i < npart; i += 256) s += partials[i];
    sh[tid] = s;
    __syncthreads();
    for (int stride = 128; stride > 0; stride >>= 1) {
        if (tid < stride) sh[tid] += sh[tid + stride];
        __syncthreads();
    }
    if (tid == 0) {
        float mean = sh[0] * invTotal;
        out[lossOff]     = 1.25f * mean;   // mean + BETA*mean, BETA = 0.25
        out[lossOff + 1] = 0.0f;           // contrastloss
    }
}

extern "C" void kernel_launch(void* const* d_in, const int* in_sizes, int n_in,
                              void* d_out, int out_size, void* d_ws, size_t ws_size,
                              hipStream_t stream) {
    const float* X  = (const float*)d_in[0];   // (N, 192) fp32
    const float* EW = (const float*)d_in[1];   // (50, 192) fp32
    float* out = (float*)d_out;
    float* ws  = (float*)d_ws;
    const int nrows = in_sizes[0] / EMB_DIM;

    vq_setup_kernel<<<1, 256, 0, stream>>>(EW, ws);
    vq_main_kernel<<<MAIN_BLOCKS, MAIN_THREADS, 0, stream>>>(X, EW, ws, out, ws + PART_OFF, nrows);

    const float invTotal = 1.0f / ((float)nrows * (float)EMB_DIM);
    const unsigned long long lossOff = (unsigned long long)nrows * EMB_DIM + (unsigned long long)nrows;
    vq_reduce_kernel<<<1, 256, 0, stream>>>(ws + PART_OFF, out, NPART, invTotal, lossOff);
}